// NeuralMDSurrogate_22351009809176
// MI455X (gfx1250) — compile-verified
//
#include <hip/hip_runtime.h>
#include <hip/hip_bf16.h>

// Problem constants (from reference): B=4, N=512, H=64
#define BB 4
#define NN 512
#define HH 64
#define MM (BB * NN)   // 2048 rows

// Pair-stage tiling: 8 i's per block (1 wave each), 64-j LDS tiles.
#define TI 8
#define TJ 64
#define NTILE (NN / TJ)   // 8

typedef __attribute__((ext_vector_type(2))) float v2f;
typedef __attribute__((ext_vector_type(4))) float v4f;
typedef __attribute__((ext_vector_type(8))) float v8f;
typedef int gv4i __attribute__((vector_size(16)));   // matches builtin param type

#if __has_builtin(__builtin_amdgcn_global_load_async_to_lds_b128)
#define HAVE_ASYNC_LDS 1
#else
#define HAVE_ASYNC_LDS 0
#endif

__device__ __forceinline__ void async_wait_all() {
#if __has_builtin(__builtin_amdgcn_s_wait_asynccnt)
  __builtin_amdgcn_s_wait_asynccnt(0);
#elif HAVE_ASYNC_LDS
  asm volatile("s_wait_asynccnt 0x0" ::: "memory");
#endif
}

// Stage a 16 KB tile (TJ*HH floats) global -> LDS; 256 threads x 4 x 16B.
__device__ __forceinline__ void stage_tile(float* ldsdst, const float* gsrc, int tid) {
#if HAVE_ASYNC_LDS
#pragma unroll
  for (int k = 0; k < 4; ++k) {
    int chunk = tid + k * 256;   // 16B chunks
    __builtin_amdgcn_global_load_async_to_lds_b128(
        (gv4i*)(gsrc + chunk * 4),
        (gv4i*)(ldsdst + chunk * 4),
        0, 0);
  }
#else
#pragma unroll
  for (int k = 0; k < 4; ++k) {
    int chunk = tid + k * 256;
    ((v4f*)ldsdst)[chunk] = ((const v4f*)gsrc)[chunk];
  }
#endif
}

// ---------------------------------------------------------------------------
// K1: first encoder layer: h1[row,c] = relu(pos_x*W[0,c] + pos_y*W[1,c] + b[c])
// ---------------------------------------------------------------------------
__global__ void k_encode1(const float* __restrict__ pos,
                          const float* __restrict__ w1,   // (2,64) row-major
                          const float* __restrict__ b1,
                          float* __restrict__ h1) {
  int idx = blockIdx.x * blockDim.x + threadIdx.x;  // MM*64 threads
  int row = idx >> 6, c = idx & 63;
  float px = pos[row * 2 + 0], py = pos[row * 2 + 1];
  float v = fmaf(px, w1[c], fmaf(py, w1[64 + c], b1[c]));
  h1[idx] = fmaxf(v, 0.0f);
}

// ---------------------------------------------------------------------------
// K2: minimum-image pairwise distances, dist[b,i,j]
// ---------------------------------------------------------------------------
__global__ void k_dist(const float* __restrict__ pos,
                       const int* __restrict__ boxp,
                       float* __restrict__ dist) {
  int idx = blockIdx.x * blockDim.x + threadIdx.x;  // B*N*N threads
  int b = idx / (NN * NN);
  int rem = idx - b * NN * NN;
  int i = rem >> 9;          // N = 512
  int j = rem & (NN - 1);
  float box = (float)boxp[0];
  float xi = pos[(b * NN + i) * 2 + 0], yi = pos[(b * NN + i) * 2 + 1];
  float xj = pos[(b * NN + j) * 2 + 0], yj = pos[(b * NN + j) * 2 + 1];
  float dx = xj - xi, dy = yj - yi;
  dx -= box * rintf(dx / box);   // rintf = round-half-even, matches jnp.round
  dy -= box * rintf(dy / box);
  dist[idx] = sqrtf(dx * dx + dy * dy);
}

// ---------------------------------------------------------------------------
// K3: fp32 WMMA GEMM: C(M x 64) = act(A(M x K) @ W(K x 64) + bias)
//   One wave = one 16x16 C tile via V_WMMA_F32_16X16X4_F32, K-loop step 4.
//   Block = 128 threads (4 waves) covering 16 rows x 64 cols. Grid = M/16.
//   Layout per ISA 7.12.2 (32-bit A 16x4 / B 4x16 / C-D 8 VGPRs):
//     lane&15 = row (A) / col (B); lane>>4 selects K pair {0,1} vs {2,3};
//     C: vgpr r = row r (lanes 0-15) / row 8+r (lanes 16-31).
// ---------------------------------------------------------------------------
template <int K, bool RELU>
__global__ void k_gemm_wmma(const float* __restrict__ A, int lda,
                            const float* __restrict__ W,     // K x 64 row-major
                            const float* __restrict__ bias,  // 64 or nullptr
                            float* __restrict__ C, int ldc) {
  int lane = threadIdx.x & 31;
  int wave = threadIdx.x >> 5;        // 0..3 -> column tile
  int row0 = blockIdx.x * 16;
  int col0 = wave * 16;
  int mn = lane & 15;
  int kh = (lane >> 4) << 1;          // 0 or 2

  v8f acc = {0.f, 0.f, 0.f, 0.f, 0.f, 0.f, 0.f, 0.f};
  const float* arow = A + (size_t)(row0 + mn) * lda + kh;   // 8B aligned
#pragma unroll
  for (int k0 = 0; k0 < K; k0 += 4) {
    v2f a = *(const v2f*)(arow + k0);
    v2f bm;
    bm.x = W[(k0 + kh + 0) * 64 + col0 + mn];
    bm.y = W[(k0 + kh + 1) * 64 + col0 + mn];
    acc = __builtin_amdgcn_wmma_f32_16x16x4_f32(
        /*neg_a=*/false, a, /*neg_b=*/false, bm,
        /*c_mod=*/(short)0, acc, /*reuse_a=*/false, /*reuse_b=*/false);
  }

  float bv = bias ? bias[col0 + mn] : 0.0f;
  int mb = (lane >> 4) * 8;
#pragma unroll
  for (int r = 0; r < 8; ++r) {
    float v = acc[r] + bv;
    if (RELU) v = fmaxf(v, 0.0f);
    C[(size_t)(row0 + mb + r) * ldc + col0 + mn] = v;
  }
}

// ---------------------------------------------------------------------------
// K4: pair reduction. For each (b,i):
//   S[b,i,:] = (1/(N-1)) * sum_{j != i} relu(ai[b,i,:] + aj[b,j,:] + d_ij*wd[:])
// Block = 256 threads (8 waves) handles TI=8 consecutive i's (wave w -> i0+w);
// each lane owns 2 channels. aj j-tiles (TJ=64 rows = 16 KB) are staged into
// LDS once per block (8x L2-traffic reduction vs per-i streaming) with
// gfx1250 async load-to-LDS, double-buffered: tile t+1 streams while tile t
// is consumed. The j==i skip is wave-uniform. No cross-wave reduction needed.
// ---------------------------------------------------------------------------
__global__ void k_pairsum(const float* __restrict__ ai,
                          const float* __restrict__ aj,
                          const float* __restrict__ dist,
                          const float* __restrict__ wd,   // 64 (msg_w1 row 2H)
                          float* __restrict__ S) {
  __shared__ alignas(16) float buf[2][TJ * HH];   // 2 x 16 KB
  int blk = blockIdx.x;                 // b * (N/TI) + itile
  int b = blk / (NN / TI);
  int i0 = (blk - b * (NN / TI)) * TI;
  int tid = threadIdx.x;
  int lane = tid & 31;
  int wv = tid >> 5;                    // 0..7 -> i = i0 + wv
  int i = i0 + wv;
  int bi = b * NN + i;
  int c = lane << 1;                    // channel base

  const float* ajb = aj + (size_t)(b * NN) * HH;
  const float* drow = dist + (size_t)bi * NN;
  v2f av = *(const v2f*)(ai + (size_t)bi * HH + c);
  v2f wv2 = *(const v2f*)(wd + c);

  stage_tile(&buf[0][0], ajb, tid);
  async_wait_all();
  __syncthreads();

  float ax = 0.0f, ay = 0.0f;
  for (int t = 0; t < NTILE; ++t) {
    if (t + 1 < NTILE)
      stage_tile(&buf[(t + 1) & 1][0], ajb + (size_t)(t + 1) * TJ * HH, tid);
    const float* lb = &buf[t & 1][0];
    int j0 = t * TJ;
#pragma unroll 4
    for (int jr = 0; jr < TJ; ++jr) {
      int j = j0 + jr;
      if (j == i) continue;                     // wave-uniform branch
      float d = drow[j];
      v2f jv = *(const v2f*)(lb + jr * HH + c); // ds_load_b64, conflict-free
      float px = fmaf(d, wv2.x, av.x + jv.x);
      float py = fmaf(d, wv2.y, av.y + jv.y);
      ax += fmaxf(px, 0.0f);
      ay += fmaxf(py, 0.0f);
    }
    async_wait_all();   // next tile landed (overlapped with compute above)
    __syncthreads();
  }
  v2f o;
  o.x = ax * (1.0f / (float)(NN - 1));
  o.y = ay * (1.0f / (float)(NN - 1));
  *(v2f*)(S + (size_t)bi * HH + c) = o;
}

// ---------------------------------------------------------------------------
// K5: final decoder head: out(M x 2) = d1(M x 64) @ dec_w2(64 x 2) + dec_b2
// ---------------------------------------------------------------------------
__global__ void k_decode(const float* __restrict__ d1,
                         const float* __restrict__ w2,  // (64,2) row-major
                         const float* __restrict__ b2,
                         float* __restrict__ out) {
  int idx = blockIdx.x * blockDim.x + threadIdx.x;  // MM*2 threads
  int row = idx >> 1, c = idx & 1;
  float s = b2[c];
  const float* dr = d1 + (size_t)row * 64;
#pragma unroll
  for (int k = 0; k < 64; ++k) s = fmaf(dr[k], w2[k * 2 + c], s);
  out[idx] = s;
}

// ---------------------------------------------------------------------------
extern "C" void kernel_launch(void* const* d_in, const int* in_sizes, int n_in,
                              void* d_out, int out_size, void* d_ws, size_t ws_size,
                              hipStream_t stream) {
  const float* pos    = (const float*)d_in[0];
  const int*   boxp   = (const int*)d_in[1];
  const float* enc_w1 = (const float*)d_in[2];
  const float* enc_b1 = (const float*)d_in[3];
  const float* enc_w2 = (const float*)d_in[4];
  const float* enc_b2 = (const float*)d_in[5];
  const float* msg_w1 = (const float*)d_in[6];   // (129, 64)
  const float* msg_b1 = (const float*)d_in[7];
  const float* msg_w2 = (const float*)d_in[8];
  const float* msg_b2 = (const float*)d_in[9];
  const float* upd_w1 = (const float*)d_in[10];  // (128, 64)
  const float* upd_b1 = (const float*)d_in[11];
  const float* upd_w2 = (const float*)d_in[12];
  const float* upd_b2 = (const float*)d_in[13];
  const float* dec_w1 = (const float*)d_in[14];
  const float* dec_b1 = (const float*)d_in[15];
  const float* dec_w2 = (const float*)d_in[16];  // (64, 2)
  const float* dec_b2 = (const float*)d_in[17];
  float* out = (float*)d_out;

  // Workspace carve-up (floats): ~7.3 MB total
  float* ws   = (float*)d_ws;
  float* dist = ws;                          // B*N*N = 1,048,576
  float* bufA = dist + (size_t)BB * NN * NN; // M*64 scratch (h1, then S)
  float* bufB = bufA + (size_t)MM * 64;      // M*64 (u1)
  float* bufC = bufB + (size_t)MM * 64;      // M*64 (ai, then u)
  float* bufD = bufC + (size_t)MM * 64;      // M*64 (aj, then d1)
  float* u_in = bufD + (size_t)MM * 64;      // M*128: [h | agg]

  // 1. h1 = relu(pos @ enc_w1 + enc_b1)
  k_encode1<<<(MM * 64) / 256, 256, 0, stream>>>(pos, enc_w1, enc_b1, bufA);
  // 2. minimum-image distance matrix
  k_dist<<<(BB * NN * NN) / 256, 256, 0, stream>>>(pos, boxp, dist);
  // 3. h = h1 @ enc_w2 + enc_b2  -> u_in[:, 0:64]   (stride 128)
  k_gemm_wmma<64, false><<<MM / 16, 128, 0, stream>>>(bufA, 64, enc_w2, enc_b2, u_in, 128);
  // 4. ai = h @ w_i + msg_b1 (bias folded)          (w_i = msg_w1 rows 0..63)
  k_gemm_wmma<64, false><<<MM / 16, 128, 0, stream>>>(u_in, 128, msg_w1, msg_b1, bufC, 64);
  // 5. aj = h @ w_j                                 (w_j = msg_w1 rows 64..127)
  k_gemm_wmma<64, false><<<MM / 16, 128, 0, stream>>>(u_in, 128, msg_w1 + 64 * 64, nullptr, bufD, 64);
  // 6. S = (1/(N-1)) * sum_{j!=i} relu(ai + aj + d*w_d)   (w_d = msg_w1 row 128)
  k_pairsum<<<BB * (NN / TI), 256, 0, stream>>>(bufC, bufD, dist, msg_w1 + 128 * 64, bufA);
  // 7. agg = S @ msg_w2 + msg_b2 -> u_in[:, 64:128] (linearity: matmul after sum)
  k_gemm_wmma<64, false><<<MM / 16, 128, 0, stream>>>(bufA, 64, msg_w2, msg_b2, u_in + 64, 128);
  // 8. u1 = relu(u_in @ upd_w1 + upd_b1)            (K = 128)
  k_gemm_wmma<128, true><<<MM / 16, 128, 0, stream>>>(u_in, 128, upd_w1, upd_b1, bufB, 64);
  // 9. u = u1 @ upd_w2 + upd_b2
  k_gemm_wmma<64, false><<<MM / 16, 128, 0, stream>>>(bufB, 64, upd_w2, upd_b2, bufC, 64);
  // 10. d1 = relu(u @ dec_w1 + dec_b1)
  k_gemm_wmma<64, true><<<MM / 16, 128, 0, stream>>>(bufC, 64, dec_w1, dec_b1, bufD, 64);
  // 11. forces = d1 @ dec_w2 + dec_b2
  k_decode<<<(MM * 2) / 256, 256, 0, stream>>>(bufD, dec_w2, dec_b2, out);
}